// PixelContrastLoss_55241869361162
// MI455X (gfx1250) — compile-verified
//
#include <hip/hip_runtime.h>
#include <hip/hip_bf16.h>

// PixelContrastLoss for MI455X (gfx1250, wave32).
// Dominant cost: one coalesced streaming pass over feats (128 MB ~ 5.8us at
// 23.3 TB/s) computing per-pixel top2-channel certainty. The only real GEMM
// (pos_ @ pos_.T, 128x128x16 fp32) uses V_WMMA_F32_16X16X4_F32.

#define B_IMG   8
#define NPIX    16384
#define NCH     256
#define NCLS    16
#define MAXVIEW 200
#define PMAX    128
#define MCAP    2048
#define SORT_SZ 2048

typedef float v2f __attribute__((ext_vector_type(2)));
typedef float v8f __attribute__((ext_vector_type(8)));

struct PlanWs {
  int   counts[B_IMG][NCLS];
  int   pair_id[B_IMG][NCLS];
  int   P, n_view, kk, D;
  int   cls_index[NCLS];
  int   pair_b[PMAX], pair_cls[PMAX], pair_cnt[PMAX], pair_stride[PMAX];
  float pos[PMAX][NCLS];   // padded rows (>=P) stay zero
  float nrm[PMAX];
  float inter_cos_sum;
  float intra_sum;
  float mask_cnt;
};

__device__ __forceinline__ int dgcd(int a, int b) {
  while (b) { int t = a % b; a = b; b = t; }
  return a;
}

// ---------------- init: zero the header (ws is poisoned to 0xAA) ------------
__global__ void pcl_init_kernel(int* w, int n) {
  for (int i = blockIdx.x * blockDim.x + threadIdx.x; i < n;
       i += gridDim.x * blockDim.x)
    w[i] = 0;
}

// ---------------- per-(image,class) histogram -------------------------------
__global__ void pcl_histo_kernel(const int* __restrict__ predict, PlanWs* ws) {
  __shared__ int hc[NCLS];
  int tid = threadIdx.x;
  if (tid < NCLS) hc[tid] = 0;
  __syncthreads();
  int gid = blockIdx.x * 256 + tid;          // N multiple of 256 -> b uniform
  int b   = gid / NPIX;
  int v   = predict[gid];
  if ((unsigned)v < NCLS) atomicAdd(&hc[v], 1);
  __syncthreads();
  if (tid < NCLS && hc[tid]) atomicAdd(&ws->counts[b][tid], hc[tid]);
}

// ---------------- build pair table, n_view, classes_index -------------------
__global__ void pcl_plan_kernel(PlanWs* ws) {
  int P = 0, nview = 0x7fffffff;
  for (int b = 0; b < B_IMG; ++b) {
    for (int c = 0; c < NCLS; ++c) {
      int cnt = ws->counts[b][c];
      ws->pair_id[b][c] = -1;
      if (cnt >= MAXVIEW) {
        int L = cnt < MCAP ? cnt : MCAP;
        ws->pair_id[b][c] = P;
        ws->pair_b[P]   = b;
        ws->pair_cls[P] = c;
        ws->pair_cnt[P] = L;
        int s = L / 2 + 1;
        while (dgcd(s, L) != 1) ++s;          // coprime-stride permutation
        ws->pair_stride[P] = s;
        if (L < nview) nview = L;
        ++P;
      }
    }
  }
  int D = 0;
  for (int c = 0; c < NCLS; ++c) {
    int present = 0;
    for (int b = 0; b < B_IMG; ++b) present |= (ws->counts[b][c] > 0);
    if (present) ws->cls_index[D++] = c;
  }
  ws->P = P;
  ws->n_view = nview;
  ws->kk = nview / 2;
  ws->D = D;
}

// ------------- ordered pixel-index gather (block prefix scan) ---------------
__global__ void pcl_gather_idx_kernel(const int* __restrict__ predict,
                                      int* __restrict__ idx_list, PlanWs* ws) {
  int b = blockIdx.x >> 4, cls = blockIdx.x & 15;
  int p = ws->pair_id[b][cls];
  if (p < 0) return;
  __shared__ int waveSum[8];
  __shared__ int base;
  int tid = threadIdx.x, lane = tid & 31, wid = tid >> 5;
  if (tid == 0) base = 0;
  __syncthreads();
  int* out = idx_list + p * MCAP;
  const int* pr = predict + b * NPIX;
  for (int start = 0; start < NPIX; start += 256) {
    int pix = start + tid;
    bool m = (pr[pix] == cls);
    unsigned long long bal = __ballot(m);
    int pre = __popcll(bal & ((1ull << lane) - 1ull));
    if (lane == 0) waveSum[wid] = __popcll(bal);
    __syncthreads();
    int woff = 0, tot = 0;
#pragma unroll
    for (int w = 0; w < 8; ++w) {
      int sv = waveSum[w];
      tot += sv;
      if (w < wid) woff += sv;
    }
    int myb = base;
    if (m) {
      int o = myb + woff + pre;
      if (o < MCAP) out[o] = pix;
    }
    __syncthreads();
    if (tid == 0) base = myb + tot;
  }
}

// -------- streaming top2-channel certainty for every pixel (128 MB pass) ----
__global__ void pcl_certainty_kernel(const float* __restrict__ feats,
                                     float* __restrict__ cert) {
  int gid  = blockIdx.x * blockDim.x + threadIdx.x;
  int base = gid * 4;                       // 4 consecutive pixels per thread
  int b    = base / NPIX;
  int pix  = base % NPIX;
  const float* fb = feats + (size_t)b * NCH * NPIX + pix;
  float t1[4], t2[4];
#pragma unroll
  for (int q = 0; q < 4; ++q) { t1[q] = -1e30f; t2[q] = -1e30f; }
  for (int c = 0; c < NCH; ++c) {
    const float4 v = *reinterpret_cast<const float4*>(fb + (size_t)c * NPIX);
    if (c + 8 < NCH) __builtin_prefetch(fb + (size_t)(c + 8) * NPIX, 0, 0);
    float vv[4] = {v.x, v.y, v.z, v.w};
#pragma unroll
    for (int q = 0; q < 4; ++q) {
      float x = vv[q];
      if (x > t1[q]) { t2[q] = t1[q]; t1[q] = x; }
      else if (x > t2[q]) t2[q] = x;
    }
  }
#pragma unroll
  for (int q = 0; q < 4; ++q) cert[base + q] = t1[q] - t2[q];
}

// ------ per-pair top-k certainty set (LDS bitonic) -> mean -> pos_, nrm -----
__global__ void pcl_topk_kernel(const float* __restrict__ feats,
                                const float* __restrict__ cert,
                                const int* __restrict__ idx_list, PlanWs* ws) {
  int p = blockIdx.x;
  if (p >= ws->P) return;
  __shared__ float skey[SORT_SZ];
  __shared__ int   sval[SORT_SZ];
  __shared__ float red[NCLS];
  __shared__ int   cdl[NCLS];
  int tid = threadIdx.x;
  int b = ws->pair_b[p], L = ws->pair_cnt[p];
  int D = ws->D, kk = ws->kk;
  const int* myIdx = idx_list + p * MCAP;
  for (int j = tid; j < SORT_SZ; j += 256) {
    if (j < L) {
      int pix = myIdx[j];
      skey[j] = cert[b * NPIX + pix];
      sval[j] = pix;
    } else {
      skey[j] = -1e30f;
      sval[j] = 0x7fffffff;
    }
  }
  if (tid < NCLS) { red[tid] = 0.f; cdl[tid] = (tid < D) ? ws->cls_index[tid] : 0; }
  // bitonic sort, descending key, ascending index on ties (top_k semantics)
  for (unsigned size = 2; size <= SORT_SZ; size <<= 1) {
    for (unsigned stride = size >> 1; stride > 0; stride >>= 1) {
      __syncthreads();
      for (unsigned i = tid; i < SORT_SZ; i += 256) {
        unsigned j2 = i ^ stride;
        if (j2 > i) {
          bool dir = ((i & size) == 0);
          float ki = skey[i], kj = skey[j2];
          int vi = sval[i], vj = sval[j2];
          bool iBefore = (ki > kj) || (ki == kj && vi < vj);
          if (dir ? !iBefore : iBefore) {
            skey[i] = kj; skey[j2] = ki;
            sval[i] = vj; sval[j2] = vi;
          }
        }
      }
    }
  }
  __syncthreads();
  // mean of top-kk pixels over the D class-indexed channels
  float acc[NCLS];
#pragma unroll
  for (int d = 0; d < NCLS; ++d) acc[d] = 0.f;
  for (int j = tid; j < kk; j += 256) {
    int pix = sval[j];
#pragma unroll
    for (int d = 0; d < NCLS; ++d)
      if (d < D)
        acc[d] += feats[((size_t)(b * NCH + cdl[d])) * NPIX + pix];
  }
#pragma unroll
  for (int d = 0; d < NCLS; ++d)
    if (d < D && acc[d] != 0.f) atomicAdd(&red[d], acc[d]);
  __syncthreads();
  if (tid < NCLS) ws->pos[p][tid] = (tid < D) ? red[tid] / (float)kk : 0.f;
  __syncthreads();
  if (tid == 0) {
    float s = 0.f;
    for (int d = 0; d < D; ++d) { float v = ws->pos[p][d]; s += v * v; }
    ws->nrm[p] = sqrtf(s);
  }
}

// --------- inter loss: cosine(selected pixel slice, pos_) accumulation ------
__global__ void pcl_inter_kernel(const float* __restrict__ feats,
                                 const int* __restrict__ idx_list, PlanWs* ws) {
  int p = blockIdx.x;
  if (p >= ws->P) return;
  __shared__ float posv[NCLS];
  __shared__ int   cdl[NCLS];
  __shared__ float pn, bsum;
  int tid = threadIdx.x;
  int D = ws->D, nv = ws->n_view;
  int b = ws->pair_b[p], L = ws->pair_cnt[p], s = ws->pair_stride[p];
  if (tid < NCLS) {
    posv[tid] = (tid < D) ? ws->pos[p][tid] : 0.f;
    cdl[tid]  = (tid < D) ? ws->cls_index[tid] : 0;
  }
  if (tid == 0) { pn = ws->nrm[p]; bsum = 0.f; }
  __syncthreads();
  const int* myIdx = idx_list + p * MCAP;
  float acc = 0.f;
  for (int r = tid; r < nv; r += 256) {
    int pix = myIdx[(s * r + p) % L];        // deterministic permutation
    float dot = 0.f, na = 0.f;
    for (int d = 0; d < D; ++d) {
      float v = feats[((size_t)(b * NCH + cdl[d])) * NPIX + pix];
      dot += v * posv[d];
      na  += v * v;
    }
    acc += dot / fmaxf(sqrtf(na) * pn, 1e-8f);
  }
  atomicAdd(&bsum, acc);
  __syncthreads();
  if (tid == 0) atomicAdd(&ws->inter_cos_sum, bsum);
}

// --------- intra loss: pos_ @ pos_^T via V_WMMA_F32_16X16X4_F32 -------------
// 8x8 grid of 16x16 tiles, one wave per tile, K=16 in four k=4 WMMA steps.
__global__ void pcl_intra_wmma_kernel(PlanWs* ws) {
  int tile = blockIdx.x;
  int i0 = (tile >> 3) * 16, j0 = (tile & 7) * 16;
  int lane = threadIdx.x;
  int m = lane & 15;
  int hiOff = (lane >= 16) ? 2 : 0;
  const float (*pos)[NCLS] = ws->pos;       // zero-padded beyond P
  v8f c = {};
#pragma unroll
  for (int kb = 0; kb < 16; kb += 4) {
    v2f a, b2;
    // A (16x4 f32): lanes 0-15 hold K=kb+{0,1}; lanes 16-31 hold K=kb+{2,3}
    a.x  = pos[i0 + m][kb + hiOff];
    a.y  = pos[i0 + m][kb + hiOff + 1];
    // B (4x16 f32) = pos_^T: same lane striping with N instead of M
    b2.x = pos[j0 + m][kb + hiOff];
    b2.y = pos[j0 + m][kb + hiOff + 1];
    c = __builtin_amdgcn_wmma_f32_16x16x4_f32(
        /*neg_a=*/false, a, /*neg_b=*/false, b2,
        /*c_mod=*/(short)0, c, /*reuse_a=*/false, /*reuse_b=*/false);
  }
  int P = ws->P;
  float s = 0.f, cnt = 0.f;
#pragma unroll
  for (int r = 0; r < 8; ++r) {             // D tile: VGPR r -> row r(+8), col m
    int row = i0 + r + ((lane >= 16) ? 8 : 0);
    int col = j0 + m;
    if (row < P && col < P && ws->pair_cls[row] != ws->pair_cls[col]) {
      float cosv = c[r] / (ws->nrm[row] * ws->nrm[col]);
      s += cosv + 1.f;
      cnt += 1.f;
    }
  }
  atomicAdd(&ws->intra_sum, s);
  atomicAdd(&ws->mask_cnt, cnt);
}

// ---------------- finalize ---------------------------------------------------
__global__ void pcl_final_kernel(PlanWs* ws, float* out) {
  float P = (float)ws->P, nv = (float)ws->n_view;
  float inter = 1.f - ws->inter_cos_sum / fmaxf(P * nv, 1.f);
  float intra = ws->intra_sum / fmaxf(ws->mask_cnt, 1.f);
  out[0] = intra + inter;
}

extern "C" void kernel_launch(void* const* d_in, const int* in_sizes, int n_in,
                              void* d_out, int out_size, void* d_ws, size_t ws_size,
                              hipStream_t stream) {
  const float* feats   = (const float*)d_in[0];
  // d_in[1] = labels (unused by the reference forward)
  const int*   predict = (const int*)d_in[2];
  float*       out     = (float*)d_out;

  PlanWs* ws = (PlanWs*)d_ws;
  size_t hdr = (sizeof(PlanWs) + 255) & ~(size_t)255;
  int*   idx_list = (int*)((char*)d_ws + hdr);
  float* cert     = (float*)(idx_list + (size_t)PMAX * MCAP);

  int hdr_words = (int)(sizeof(PlanWs) / sizeof(int));
  pcl_init_kernel<<<16, 256, 0, stream>>>((int*)d_ws, hdr_words);
  pcl_histo_kernel<<<(B_IMG * NPIX) / 256, 256, 0, stream>>>(predict, ws);
  pcl_plan_kernel<<<1, 1, 0, stream>>>(ws);
  pcl_gather_idx_kernel<<<B_IMG * NCLS, 256, 0, stream>>>(predict, idx_list, ws);
  pcl_certainty_kernel<<<(B_IMG * NPIX) / 1024, 256, 0, stream>>>(feats, cert);
  pcl_topk_kernel<<<PMAX, 256, 0, stream>>>(feats, cert, idx_list, ws);
  pcl_inter_kernel<<<PMAX, 256, 0, stream>>>(feats, idx_list, ws);
  pcl_intra_wmma_kernel<<<64, 32, 0, stream>>>(ws);
  pcl_final_kernel<<<1, 1, 0, stream>>>(ws, out);
}